// FoldNd_60782377173029
// MI455X (gfx1250) — compile-verified
//
#include <hip/hip_runtime.h>

// FoldNd (col2im) for MI455X / gfx1250.
//
// Reference: x[32, 64*3*3, 4096] fp32, scatter-add into [32,64,66,66] then crop
// to [32,64,64,64]. Rewritten as a gather (each input element lands in exactly
// one output), so no atomics:
//   out[b,c,oh,ow] = sum_{kh,kw} x[b, c*9+kh*3+kw, (oh+1-kh)*64 + (ow+1-kw)]
//                    (terms with bh/bw outside [0,64) are zero)
//
// Memory-bound: ~302MB read + ~33.5MB write => ~14.4us floor @ 23.3 TB/s.
// The 9 kernel planes for a channel are contiguous (c*9+k), so each block
// stages its (b,c,row-tile) working set into LDS with ONE Tensor-Data-Mover
// DMA (tensor_load_to_lds): scalar-issued, perfectly coalesced, async
// (TENSORcnt). The +-1 shifted reads then hit LDS where unaligned access is
// free, and stores are coalesced float4.

typedef unsigned int v4u __attribute__((ext_vector_type(4)));
typedef int          v4i __attribute__((ext_vector_type(4)));
typedef int          v8i __attribute__((ext_vector_type(8)));

#define KK        9          // 3x3 kernel offsets
#define PLANE     4096       // 64*64 floats per (b, ck) plane
#define ROWS_MAX  18         // rows of each plane staged per 16-row tile

__global__ __launch_bounds__(256)
void fold_tdm_kernel(const float* __restrict__ x, float* __restrict__ out) {
    __shared__ float smem[KK * ROWS_MAX * 64];   // 41472 bytes

    const int tile = blockIdx.x;          // 0..8191
    const int bc   = tile >> 2;           // b*64 + c  (0..2047)
    const int ts   = (tile & 3) << 4;     // output-row tile start: 0,16,32,48

    // Input rows needed: bh = oh+1-kh for oh in [ts, ts+15], kh in {0,1,2}
    //   => bh in [ts-1, ts+16], clipped to the 64-row plane.
    const int loadStart   = (ts == 0)  ? 0  : ts - 1;
    const int loadEnd     = (ts == 48) ? 63 : ts + 16;
    const int nrows       = loadEnd - loadStart + 1;   // 17 or 18
    const int planeStride = nrows * 64;                // LDS floats per plane

#if __has_builtin(__builtin_amdgcn_tensor_load_to_lds)
    if (threadIdx.x < 32) {   // wave 0 issues the DMA (TDM ignores EXEC)
        const unsigned long long ga =
            (unsigned long long)(const void*)x
            + (unsigned long long)bc * (unsigned long long)(KK * PLANE * 4)
            + (unsigned long long)loadStart * 256ull;          // 64 floats/row
        const unsigned int ldsOff = (unsigned int)(size_t)(&smem[0]);

        // ---- D# group 0: count / lds_addr / global_addr / type=2 ----
        v4u g0;
        g0[0] = 1u;                                            // count=1 (valid user D#)
        g0[1] = ldsOff;                                        // lds_addr (bytes)
        g0[2] = (unsigned int)(ga & 0xFFFFFFFFull);            // global_addr[31:0]
        g0[3] = (unsigned int)((ga >> 32) & 0x1FFFFFFull)      // global_addr[56:32]
              | (2u << 30);                                    // type = 2 ("image")

        // ---- D# group 1: data_size, dims, tile dims, strides ----
        // 3D tile: X=64 cols, Y=nrows rows (stride 64), Z=9 planes (stride 4096)
        v8i g1;
        g1[0] = (2 << 16);                 // data_size=2 (4B); mask/pad/iterate=0
        g1[1] = (64 << 16);                // tensor_dim0[15:0] = 64
        g1[2] = (nrows << 16);             // tensor_dim0[31:16]=0 | tensor_dim1[15:0]=nrows
        g1[3] = (64 << 16);                // tensor_dim1[31:16]=0 | tile_dim0=64
        g1[4] = nrows | (KK << 16);        // tile_dim1=nrows | tile_dim2=9
        g1[5] = 64;                        // tensor_dim0_stride[31:0] = 64
        g1[6] = (PLANE << 16);             // stride0[47:32]=0 | tensor_dim1_stride[15:0]=4096
        g1[7] = 0;                         // tensor_dim1_stride[47:16] = 0

        // ---- D# group 2: tensor_dim2=9 (exact => no OOB), rest unused ----
        v4i g2; g2[0] = KK; g2[1] = 0; g2[2] = 0; g2[3] = 0;
        v4i g3; g3[0] = 0;  g3[1] = 0; g3[2] = 0; g3[3] = 0;
        // 5th operand (clang-23 6-arg form): ISA VADDR4 slot is unused -> zeros.
        v8i g4; g4[0] = 0; g4[1] = 0; g4[2] = 0; g4[3] = 0;
                g4[4] = 0; g4[5] = 0; g4[6] = 0; g4[7] = 0;

        __builtin_amdgcn_tensor_load_to_lds(g0, g1, g2, g3, g4, 0);
        __builtin_amdgcn_s_wait_tensorcnt(0);
    }
#else
    // Fallback: cooperative global->LDS staging (identical LDS layout).
    {
        const float* src = x + (size_t)bc * (KK * PLANE) + (size_t)loadStart * 64;
        const int total = KK * planeStride;
        for (int i = threadIdx.x; i < total; i += 256) {
            const int k   = i / planeStride;
            const int rem = i - k * planeStride;
            smem[i] = src[(size_t)k * PLANE + rem];
        }
    }
#endif
    __syncthreads();

    // ---- Compute: 1024 outputs per tile, one float4 per thread ----
    const int t    = threadIdx.x;
    const int oh   = ts + (t >> 4);       // output row
    const int col4 = (t & 15) << 2;       // output col base (multiple of 4)

    float acc[4] = {0.f, 0.f, 0.f, 0.f};

#pragma unroll
    for (int kh = 0; kh < 3; ++kh) {
        const int bh = oh + 1 - kh;       // source row in plane
        if (bh < loadStart || bh > loadEnd) continue;   // == bh in [0,64)
        const int rowBase = (bh - loadStart) * 64;
#pragma unroll
        for (int kw = 0; kw < 3; ++kw) {
            const float* p = &smem[(kh * 3 + kw) * planeStride + rowBase];
            const int shift = 1 - kw;     // bw = ow + shift
#pragma unroll
            for (int j = 0; j < 4; ++j) {
                const int bw = col4 + j + shift;
                if (bw >= 0 && bw < 64) acc[j] += p[bw];
            }
        }
    }

    const size_t obase = (size_t)bc * PLANE + (size_t)oh * 64 + (size_t)col4;
    float4 o; o.x = acc[0]; o.y = acc[1]; o.z = acc[2]; o.w = acc[3];
    *(float4*)(out + obase) = o;
}

extern "C" void kernel_launch(void* const* d_in, const int* in_sizes, int n_in,
                              void* d_out, int out_size, void* d_ws, size_t ws_size,
                              hipStream_t stream) {
    const float* x   = (const float*)d_in[0];
    float*       out = (float*)d_out;
    (void)in_sizes; (void)n_in; (void)out_size; (void)d_ws; (void)ws_size;

    // 32 batches * 64 channels * 4 row-tiles = 8192 blocks, 256 threads each.
    fold_tdm_kernel<<<dim3(8192), dim3(256), 0, stream>>>(x, out);
}